// RoiPoolingContrastive_25950192402928
// MI455X (gfx1250) — compile-verified
//
#include <hip/hip_runtime.h>
#include <hip/hip_bf16.h>

// ---------------------------------------------------------------------------
// RoiPooling + contrastive fg-score for MI455X (gfx1250, wave32).
//
// Strategy: 2D integral image (chunked over channels, 32 at a time) so the
// pooling is 4 loads per (box, channel).  The row/column prefix scans are done
// on the matrix pipe: prefix-sum of a 16x16 tile == multiply by a triangular
// ones-matrix, i.e. 4 chained V_WMMA_F32_16X16X4_F32 ops, with cross-tile
// carries propagated by wave32 shuffles.
// ---------------------------------------------------------------------------

typedef __attribute__((ext_vector_type(2))) float v2f;
typedef __attribute__((ext_vector_type(8))) float v8f;

#define HH  512
#define WW  512
#define SH  513          // integral image rows  (H+1)
#define SWS 520          // padded row stride for integral image (>= W+1)
#define CC  32           // channels per chunk

// ---------------------------------------------------------------------------
// 1) argmax over K class maps -> predicted class per pixel
// ---------------------------------------------------------------------------
__global__ __launch_bounds__(256)
void cam_argmax_kernel(const float* __restrict__ cam_bg, int* __restrict__ cam, int K) {
    const int HW = HH * WW;
    int p = blockIdx.x * blockDim.x + threadIdx.x;
    if (p >= HW) return;
    float best = cam_bg[p];
    int   bk   = 0;
    for (int k = 1; k < K; ++k) {
        if (k + 1 < K) __builtin_prefetch(cam_bg + (size_t)(k + 1) * HW + p, 0, 0);
        float v = cam_bg[(size_t)k * HW + p];
        if (v > best) { best = v; bk = k; }   // strict > keeps first max (jnp.argmax)
    }
    cam[p] = bk;
}

// ---------------------------------------------------------------------------
// 2) fg_score: one wave32 per box, count pixels whose class == label
// ---------------------------------------------------------------------------
__global__ __launch_bounds__(32)
void fg_score_kernel(const int* __restrict__ cam, const int* __restrict__ boxes,
                     const int* __restrict__ labels, float* __restrict__ out_fg) {
    const int n    = blockIdx.x;
    const int lane = threadIdx.x;
    const int x0 = boxes[n * 4 + 0], y0 = boxes[n * 4 + 1];
    const int x1 = boxes[n * 4 + 2], y1 = boxes[n * 4 + 3];
    const int label = labels[n];

    int count = 0;
    for (int y = y0; y < y1; ++y) {
        const int* row = cam + (size_t)y * WW;
        for (int x = x0 + lane; x < x1; x += 32)
            count += (row[x] == label) ? 1 : 0;
    }
    for (int off = 16; off > 0; off >>= 1)
        count += __shfl_xor(count, off, 32);
    if (lane == 0) {
        float area = (float)((x1 - x0) * (y1 - y0));
        out_fg[n] = (float)count / area;
    }
}

// ---------------------------------------------------------------------------
// 3) zero the top row and left column of each per-channel integral image
// ---------------------------------------------------------------------------
__global__ void zero_borders_kernel(float* __restrict__ S) {
    const int per = SH + (SH - 1);               // 513 top-row + 512 left-col
    int t = blockIdx.x * blockDim.x + threadIdx.x;
    int c = t / per;
    int r = t % per;
    if (c >= CC) return;
    float* Sc = S + (size_t)c * SH * SWS;
    if (r < SH) Sc[r] = 0.0f;                              // row 0
    else        Sc[(size_t)(r - SH + 1) * SWS] = 0.0f;     // col 0, rows 1..512
}

// ---------------------------------------------------------------------------
// 4) row scan (inclusive prefix along x) via WMMA: P = T * U  (U upper-tri 1s)
//    One wave32 per (channel, 16-row slab); walks 32 tiles left->right.
//    f32 WMMA layouts (ISA 7.12.2):
//      A 16x4 : lane L holds row L%16, K = j + 2*(L/16)   (j = component)
//      B  4x16: lane L holds col L%16, K = j + 2*(L/16)
//      C/D    : lane L, VGPR v holds row v + 8*(L/16), col L%16
//    Key codegen detail: each lane loads exactly the two A elements it owns
//    as a float2 at a *computed address* (hi folded into the offset), so no
//    dynamic register-array indexing / cndmask chains are generated.
// ---------------------------------------------------------------------------
__global__ __launch_bounds__(32)
void row_scan_wmma_kernel(const float* __restrict__ F, float* __restrict__ S,
                          int chunkBase) {
    const int lane = threadIdx.x;
    const int slabsPerC = HH / 16;                 // 32
    const int cidx = blockIdx.x / slabsPerC;
    const int slab = blockIdx.x % slabsPerC;
    const int cg   = chunkBase + cidx;

    const int hi = lane >> 4;                      // lane group (0/1)
    const int lo = lane & 15;

    const float* Fb = F + (size_t)cg * HH * WW + (size_t)slab * 16 * WW
                        + (size_t)lo * WW;         // this lane's row
    float*       Sb = S + (size_t)cidx * SH * SWS + (size_t)(slab * 16 + 1) * SWS;

    // B = upper-triangular ones, built once (loop-invariant)
    v2f bu[4];
    #pragma unroll
    for (int s = 0; s < 4; ++s) {
        const int k0 = 4 * s + 2 * hi;
        bu[s].x = (k0     <= lo) ? 1.0f : 0.0f;
        bu[s].y = (k0 + 1 <= lo) ? 1.0f : 0.0f;
    }

    float carry[8];
    #pragma unroll
    for (int v = 0; v < 8; ++v) carry[v] = 0.0f;

    for (int t = 0; t < WW / 16; ++t) {
        const float2* rp2 = (const float2*)(Fb + t * 16);
        v8f acc = {};
        #pragma unroll
        for (int s = 0; s < 4; ++s) {
            float2 q = rp2[2 * s + hi];            // A[M=lo, K=4s+2hi .. +1]
            v2f a;  a.x = q.x;  a.y = q.y;
            acc = __builtin_amdgcn_wmma_f32_16x16x4_f32(
                      false, a, false, bu[s], (short)0, acc, false, false);
        }
        // acc[v] = prefix of row (v + 8*hi) at column lo; col-15 = segment total
        #pragma unroll
        for (int v = 0; v < 8; ++v) {
            float segTot = __shfl(acc[v], lane | 15, 32);
            Sb[(size_t)(v + 8 * hi) * SWS + t * 16 + lo + 1] = acc[v] + carry[v];
            carry[v] += segTot;
        }
    }
}

// ---------------------------------------------------------------------------
// 5) column scan (inclusive prefix along y) via WMMA: P = L * T (L lower-tri 1s)
//    One wave32 per (channel, 16-column strip); walks 32 tiles top->bottom,
//    updating the integral image in place.
// ---------------------------------------------------------------------------
__global__ __launch_bounds__(32)
void col_scan_wmma_kernel(float* __restrict__ S) {
    const int lane = threadIdx.x;
    const int stripsPerC = WW / 16;                // 32
    const int cidx  = blockIdx.x / stripsPerC;
    const int strip = blockIdx.x % stripsPerC;

    float*    Sc = S + (size_t)cidx * SH * SWS;
    const int xb = strip * 16 + 1;
    const int hi = lane >> 4;
    const int lo = lane & 15;

    // A = lower-triangular ones, built once (loop-invariant)
    v2f al[4];
    #pragma unroll
    for (int s = 0; s < 4; ++s) {
        const int k0 = 4 * s + 2 * hi;
        al[s].x = (k0     <= lo) ? 1.0f : 0.0f;
        al[s].y = (k0 + 1 <= lo) ? 1.0f : 0.0f;
    }

    float carry = 0.0f;                            // running sum of column xb+lo

    for (int yt = 0; yt < HH / 16; ++yt) {
        const int yb = yt * 16 + 1;
        v8f acc = {};
        #pragma unroll
        for (int s = 0; s < 4; ++s) {
            const int k0 = 4 * s + 2 * hi;
            v2f b;                                  // B = tile[K, lo]
            b.x = Sc[(size_t)(yb + k0    ) * SWS + xb + lo];
            b.y = Sc[(size_t)(yb + k0 + 1) * SWS + xb + lo];
            acc = __builtin_amdgcn_wmma_f32_16x16x4_f32(
                      false, al[s], false, b, (short)0, acc, false, false);
        }
        // column total = D[15, lo], held by lane 16+lo in VGPR 7
        float colTot = __shfl(acc[7], 16 + lo, 32);
        #pragma unroll
        for (int v = 0; v < 8; ++v)
            Sc[(size_t)(yb + v + 8 * hi) * SWS + xb + lo] = acc[v] + carry;
        carry += colTot;
    }
}

// ---------------------------------------------------------------------------
// 6) gather: pool[n, base+c] = 4-point integral lookup / area
// ---------------------------------------------------------------------------
__global__ __launch_bounds__(32)
void pool_gather_kernel(const float* __restrict__ S, const int* __restrict__ boxes,
                        float* __restrict__ out, int C, int chunkBase) {
    const int n = blockIdx.x;
    const int c = threadIdx.x;                     // 0..CC-1
    const int x0 = boxes[n * 4 + 0], y0 = boxes[n * 4 + 1];
    const int x1 = boxes[n * 4 + 2], y1 = boxes[n * 4 + 3];
    const float area = (float)((x1 - x0) * (y1 - y0));
    const float* Sc = S + (size_t)c * SH * SWS;
    float s = Sc[(size_t)y1 * SWS + x1] - Sc[(size_t)y0 * SWS + x1]
            - Sc[(size_t)y1 * SWS + x0] + Sc[(size_t)y0 * SWS + x0];
    out[(size_t)n * C + chunkBase + c] = s / area;
}

// ---------------------------------------------------------------------------
extern "C" void kernel_launch(void* const* d_in, const int* in_sizes, int n_in,
                              void* d_out, int out_size, void* d_ws, size_t ws_size,
                              hipStream_t stream) {
    (void)n_in; (void)out_size; (void)ws_size;

    const float* F      = (const float*)d_in[0];   // (C,H,W) f32
    const float* camBg  = (const float*)d_in[1];   // (K,H,W) f32
    const int*   boxes  = (const int*)d_in[2];     // (N,4)   i32
    const int*   labels = (const int*)d_in[3];     // (N,)    i32
    float*       out    = (float*)d_out;           // pool (N*C) ++ fg_score (N)

    const int HW = HH * WW;
    const int C  = in_sizes[0] / HW;               // 256
    const int K  = in_sizes[1] / HW;               // 21
    const int N  = in_sizes[3];                    // 2560

    // workspace: [ cam_predict : HW ints | S : CC * SH * SWS floats ]
    int*   cam      = (int*)d_ws;
    size_t camBytes = (((size_t)HW * sizeof(int)) + 255) & ~(size_t)255;
    float* S        = (float*)((char*)d_ws + camBytes);

    cam_argmax_kernel<<<(HW + 255) / 256, 256, 0, stream>>>(camBg, cam, K);
    fg_score_kernel<<<N, 32, 0, stream>>>(cam, boxes, labels, out + (size_t)N * C);

    const int nChunks = C / CC;                    // 8
    const int zbTotal = CC * (SH + SH - 1);
    for (int ch = 0; ch < nChunks; ++ch) {
        const int base = ch * CC;
        zero_borders_kernel<<<(zbTotal + 255) / 256, 256, 0, stream>>>(S);
        row_scan_wmma_kernel<<<CC * (HH / 16), 32, 0, stream>>>(F, S, base);
        col_scan_wmma_kernel<<<CC * (WW / 16), 32, 0, stream>>>(S);
        pool_gather_kernel<<<N, 32, 0, stream>>>(S, boxes, out, C, base);
    }
}